// HeteroConvModel_57827439674002
// MI455X (gfx1250) — compile-verified
//
#include <hip/hip_runtime.h>
#include <hip/hip_bf16.h>

typedef float v2f __attribute__((ext_vector_type(2)));
typedef float v8f __attribute__((ext_vector_type(8)));

#define NP 200000
#define NA 100000

// ---------------------------------------------------------------- zero
__global__ void zero_f32(float* __restrict__ p, long n) {
    long i = (long)blockIdx.x * blockDim.x + threadIdx.x;
    if (i < n) p[i] = 0.0f;
}

// ---------------------------------------------------------------- degree / count
__global__ void count_kernel(const int* __restrict__ dst, int E, float* __restrict__ acc) {
    int e = blockIdx.x * blockDim.x + threadIdx.x;
    if (e < E) atomicAdd(&acc[dst[e]], 1.0f);
}

// deg -> dinv = rsqrt(deg + 1) in place (self loop adds 1)
__global__ void rsqrt_kernel(float* __restrict__ deg, int n) {
    int i = blockIdx.x * blockDim.x + threadIdx.x;
    if (i < n) deg[i] = rsqrtf(deg[i] + 1.0f);
}

// ---------------------------------------------------------------- WMMA GEMM
// Y[M,32] = X[M,128] @ W[128,32], one wave per 16-row tile, two 16x16 N-tiles,
// K unrolled in steps of 4 using V_WMMA_F32_16X16X4_F32 (exact f32 math).
__global__ void gemm_k128_n32(const float* __restrict__ X,
                              const float* __restrict__ W,
                              float* __restrict__ Y, int M) {
    const int lane = threadIdx.x & 31;
    const int wave = threadIdx.x >> 5;
    const int tile = blockIdx.x * (blockDim.x >> 5) + wave;
    const int row0 = tile * 16;
    if (row0 >= M) return;

    const int m    = lane & 15;        // row within tile (A) / col within tile (B,D)
    const int half = lane >> 4;        // lane half selects K pair / M half
    const float* __restrict__ xrow = X + (size_t)(row0 + m) * 128;

    v8f c0 = {0.f,0.f,0.f,0.f,0.f,0.f,0.f,0.f};
    v8f c1 = {0.f,0.f,0.f,0.f,0.f,0.f,0.f,0.f};

#pragma unroll
    for (int kb = 0; kb < 128; kb += 4) {
        const int krow = kb + 2 * half;
        v2f a, b0, b1;
        // A 16x4: VGPR0 -> K = 2*half, VGPR1 -> K = 2*half+1
        a.x = xrow[krow + 0];
        a.y = xrow[krow + 1];
        // B 4x16 per n-tile: VGPR0 lanes hold row K=2*half, VGPR1 row K=2*half+1
        b0.x = W[(size_t)krow * 32 + m];
        b0.y = W[(size_t)(krow + 1) * 32 + m];
        b1.x = W[(size_t)krow * 32 + 16 + m];
        b1.y = W[(size_t)(krow + 1) * 32 + 16 + m];
        c0 = __builtin_amdgcn_wmma_f32_16x16x4_f32(false, a, false, b0, (short)0, c0, false, false);
        c1 = __builtin_amdgcn_wmma_f32_16x16x4_f32(false, a, false, b1, (short)0, c1, false, false);
    }

    // D layout: VGPR v -> M = v + 8*half, N = lane&15
#pragma unroll
    for (int v = 0; v < 8; ++v) {
        const int rr = row0 + v + 8 * half;
        Y[(size_t)rr * 32 + m]      = c0[v];
        Y[(size_t)rr * 32 + 16 + m] = c1[v];
    }
}

// ---------------------------------------------------------------- GCN edge scatter
// one thread per (edge, feature): acc[dst][f] += h[src][f] * dinv[src]*dinv[dst]
__global__ void gcn_scatter(const int* __restrict__ src, const int* __restrict__ dst,
                            const float* __restrict__ h, const float* __restrict__ dinv,
                            float* __restrict__ acc, int nWork) {
    int t = blockIdx.x * blockDim.x + threadIdx.x;
    if (t >= nWork) return;
    const int e = t >> 5, f = t & 31;
    const int s = src[e], d = dst[e];
    const float w = dinv[s] * dinv[d];
    atomicAdd(&acc[(size_t)d * 32 + f], h[(size_t)s * 32 + f] * w);
}

// ---------------------------------------------------------------- SAGE edge scatter
__global__ void sage_scatter(const int* __restrict__ src, const int* __restrict__ dst,
                             const float* __restrict__ xa, float* __restrict__ acc,
                             int nWork) {
    int t = blockIdx.x * blockDim.x + threadIdx.x;
    if (t >= nWork) return;
    const int e = t >> 5, f = t & 31;
    const int s = src[e], d = dst[e];
    atomicAdd(&acc[(size_t)d * 32 + f], xa[(size_t)s * 32 + f]);
}

// ---------------------------------------------------------------- finalize + head
// one wave per paper node: combine GCN + SAGE terms, dot with W_out via wave32 shuffle.
__global__ void finalize_kernel(const float* __restrict__ acc_gcn,
                                const float* __restrict__ h,
                                const float* __restrict__ dinv,
                                const float* __restrict__ b_gcn,
                                const float* __restrict__ acc_sage,
                                const float* __restrict__ cnt,
                                const float* __restrict__ b_sage_l,
                                const float* __restrict__ xr,
                                const float* __restrict__ W_out,
                                const float* __restrict__ b_out,
                                float* __restrict__ out, int N) {
    int t = blockIdx.x * blockDim.x + threadIdx.x;
    const int i = t >> 5, f = t & 31;
    if (i >= N) return;
    const size_t base = (size_t)i * 32 + f;
    const float di = dinv[i];
    const float g  = acc_gcn[base] + h[base] * di * di + b_gcn[f];
    const float s  = acc_sage[base] / fmaxf(cnt[i], 1.0f) + b_sage_l[f] + xr[base];
    float v = (g + s) * W_out[f];
#pragma unroll
    for (int o = 16; o > 0; o >>= 1) v += __shfl_down(v, o, 32);
    if (f == 0) out[i] = v + b_out[0];
}

// ---------------------------------------------------------------- launch
extern "C" void kernel_launch(void* const* d_in, const int* in_sizes, int n_in,
                              void* d_out, int out_size, void* d_ws, size_t ws_size,
                              hipStream_t stream) {
    const float* x_paper  = (const float*)d_in[0];
    const float* x_author = (const float*)d_in[1];
    const int*   cites    = (const int*)d_in[2];
    const int*   wsrc     = (const int*)d_in[3];
    const int*   wdst     = (const int*)d_in[4];
    const float* W_gcn    = (const float*)d_in[5];
    const float* b_gcn    = (const float*)d_in[6];
    const float* W_sage_l = (const float*)d_in[7];
    const float* b_sage_l = (const float*)d_in[8];
    const float* W_sage_r = (const float*)d_in[9];
    const float* W_out    = (const float*)d_in[10];
    const float* b_out    = (const float*)d_in[11];

    const int EC = in_sizes[2] / 2;   // 6.4M cites edges
    const int EW = in_sizes[3];       // 2M  writes edges
    const int* csrc = cites;
    const int* cdst = cites + EC;

    // workspace layout (floats)
    float* ws   = (float*)d_ws;
    float* deg  = ws;                              // NP       (becomes dinv)
    float* cnt  = deg  + NP;                       // NP
    float* h    = cnt  + NP;                       // NP*32
    float* xr   = h    + (size_t)NP * 32;          // NP*32
    float* xa   = xr   + (size_t)NP * 32;          // NA*32
    float* accg = xa   + (size_t)NA * 32;          // NP*32
    float* accs = accg + (size_t)NP * 32;          // NP*32

    const int B = 256;

    // zero deg+cnt (contiguous) and both accumulators (contiguous)
    {
        long n1 = 2L * NP;
        zero_f32<<<(int)((n1 + B - 1) / B), B, 0, stream>>>(deg, n1);
        long n2 = 2L * NP * 32;
        zero_f32<<<(int)((n2 + B - 1) / B), B, 0, stream>>>(accg, n2);
    }

    // degree(cites dst) and writes count
    count_kernel<<<(EC + B - 1) / B, B, 0, stream>>>(cdst, EC, deg);
    count_kernel<<<(EW + B - 1) / B, B, 0, stream>>>(wdst, EW, cnt);
    rsqrt_kernel<<<(NP + B - 1) / B, B, 0, stream>>>(deg, NP);

    // three WMMA GEMMs (8 waves/block -> 128 rows per block)
    {
        int tilesP = NP / 16, tilesA = NA / 16, wpb = B / 32;
        gemm_k128_n32<<<(tilesP + wpb - 1) / wpb, B, 0, stream>>>(x_paper,  W_gcn,    h,  NP);
        gemm_k128_n32<<<(tilesP + wpb - 1) / wpb, B, 0, stream>>>(x_paper,  W_sage_r, xr, NP);
        gemm_k128_n32<<<(tilesA + wpb - 1) / wpb, B, 0, stream>>>(x_author, W_sage_l, xa, NA);
    }

    // edge scatters at width 32
    {
        int nWork = EC * 32;
        gcn_scatter<<<(nWork + B - 1) / B, B, 0, stream>>>(csrc, cdst, h, deg, accg, nWork);
    }
    {
        int nWork = EW * 32;
        sage_scatter<<<(nWork + B - 1) / B, B, 0, stream>>>(wsrc, wdst, xa, accs, nWork);
    }

    // finalize + linear head
    {
        int nWork = NP * 32;
        finalize_kernel<<<(nWork + B - 1) / B, B, 0, stream>>>(
            accg, h, deg, b_gcn, accs, cnt, b_sage_l, xr, W_out, b_out,
            (float*)d_out, NP);
    }
}